// GraphConv_9672266350627
// MI455X (gfx1250) — compile-verified
//
#include <hip/hip_runtime.h>

#define N_NODES 100000
#define N_EDGES 1000000
#define D_IN    64
#define D_HID   256
#define D_OUT   64
#define MT      32          // nodes per block (2 x 16-row WMMA subtiles)

typedef __attribute__((ext_vector_type(2))) float v2f;
typedef __attribute__((ext_vector_type(8))) float v8f;

// No-return fp32 global atomic add (STOREcnt-tracked, guaranteed HW atomic,
// never a CAS loop). gfx1250 VGLOBAL encoding: vaddr(64b pair), vdata, off.
__device__ __forceinline__ void atomic_add_f32_nr(float* p, float v) {
    asm volatile("global_atomic_add_f32 %0, %1, off" : : "v"(p), "v"(v) : "memory");
}

// ---------------------------------------------------------------------------
// Kernel 1: zero the aggregation buffer (N_NODES x D_IN fp32)
// ---------------------------------------------------------------------------
__global__ __launch_bounds__(256) void gin_zero(float* __restrict__ agg) {
    size_t t = (size_t)blockIdx.x * 256 + threadIdx.x;     // 1.6M threads
    float4 z = make_float4(0.f, 0.f, 0.f, 0.f);
    *(float4*)(agg + t * 4) = z;
}

// ---------------------------------------------------------------------------
// Kernel 2: scatter-add  agg[dst] += x[src]  (16 threads per edge, float4 each)
// ---------------------------------------------------------------------------
__global__ __launch_bounds__(256) void gin_scatter(const float* __restrict__ x,
                                                   const long long* __restrict__ ei,
                                                   float* __restrict__ agg) {
    size_t t = (size_t)blockIdx.x * 256 + threadIdx.x;     // 16M threads
    int e = (int)(t >> 4);
    int f = (int)(t & 15) * 4;
    long long src = ei[e];
    long long dst = ei[(size_t)N_EDGES + e];
    float4 v = *(const float4*)(x + (size_t)src * D_IN + f);
    float* p = agg + (size_t)dst * D_IN + f;
    atomic_add_f32_nr(p + 0, v.x);
    atomic_add_f32_nr(p + 1, v.y);
    atomic_add_f32_nr(p + 2, v.z);
    atomic_add_f32_nr(p + 3, v.w);
}

// ---------------------------------------------------------------------------
// Kernel 3: fused  out = relu((x+agg) @ W1 + b1) @ W2 + b2
// One block (8 wave32) per 32-node tile. fp32 WMMA 16x16x4.
// Each wave loads one B fragment per K-step and reuses it across both
// 16-row M-subtiles (halves VMEM per WMMA vs. a 16-node tile).
// ---------------------------------------------------------------------------
__global__ __launch_bounds__(256) void gin_mlp(const float* __restrict__ x,
                                               const float* __restrict__ agg,
                                               const float* __restrict__ W1,
                                               const float* __restrict__ b1,
                                               const float* __restrict__ W2,
                                               const float* __restrict__ b2,
                                               float* __restrict__ out) {
    __shared__ float sh_h[MT][D_IN + 4];      // h = x + agg        (~8.7 KB)
    __shared__ float sh_hid[MT][D_HID + 4];   // relu(h@W1+b1)      (~33.3 KB)
    __shared__ float sh_part[2][MT][D_OUT];   // GEMM2 K-half parts (16 KB)

    const int tid   = threadIdx.x;
    const int wave  = tid >> 5;               // 0..7
    const int lane  = tid & 31;
    const int half  = lane >> 4;              // 0: lanes 0-15, 1: lanes 16-31
    const int l16   = lane & 15;
    const int node0 = blockIdx.x * MT;

    // Warm L0 for this block's weight reads (global_prefetch_b8).
    __builtin_prefetch(W1 + tid * 64, 0, 3);  // 256 threads x 256B = all of W1
    __builtin_prefetch(W2 + tid * 64, 0, 3);  // 256 threads x 256B = all of W2

    // ---- Stage 0: h = x + agg  (32 x 64 = 2048 floats, 8 per thread) ----
    {
        int i = tid * 8;
        int m = i >> 6, f = i & 63;
        size_t off = (size_t)(node0 + m) * D_IN + f;
        float4 xv0 = *(const float4*)(x + off);
        float4 av0 = *(const float4*)(agg + off);
        float4 xv1 = *(const float4*)(x + off + 4);
        float4 av1 = *(const float4*)(agg + off + 4);
        sh_h[m][f + 0] = xv0.x + av0.x;
        sh_h[m][f + 1] = xv0.y + av0.y;
        sh_h[m][f + 2] = xv0.z + av0.z;
        sh_h[m][f + 3] = xv0.w + av0.w;
        sh_h[m][f + 4] = xv1.x + av1.x;
        sh_h[m][f + 5] = xv1.y + av1.y;
        sh_h[m][f + 6] = xv1.z + av1.z;
        sh_h[m][f + 7] = xv1.w + av1.w;
    }
    __syncthreads();

    // ---- GEMM1: hidden[32x256] = h[32x64] @ W1[64x256], +b1, ReLU ----
    // Wave owns two 16-wide hidden column tiles; B fragment shared by both
    // M-subtiles. A frag (16x4 f32): lane half selects K pair; VGPR idx = K.
    #pragma unroll
    for (int nt = 0; nt < 2; ++nt) {
        const int ncol = (wave * 2 + nt) * 16 + l16;   // hidden column
        v8f acc0 = {}, acc1 = {};
        #pragma unroll 4
        for (int k0 = 0; k0 < D_IN; k0 += 4) {
            const int kb = k0 + half * 2;
            v2f b;
            b.x = W1[(size_t)(kb + 0) * D_HID + ncol];
            b.y = W1[(size_t)(kb + 1) * D_HID + ncol];
            v2f a0, a1;
            a0.x = sh_h[l16][kb + 0];
            a0.y = sh_h[l16][kb + 1];
            a1.x = sh_h[16 + l16][kb + 0];
            a1.y = sh_h[16 + l16][kb + 1];
            acc0 = __builtin_amdgcn_wmma_f32_16x16x4_f32(
                false, a0, false, b, (short)0, acc0, false, false);
            acc1 = __builtin_amdgcn_wmma_f32_16x16x4_f32(
                false, a1, false, b, (short)0, acc1, false, false);
        }
        const float bias = b1[ncol];
        #pragma unroll
        for (int r = 0; r < 8; ++r) {
            float v0 = acc0[r] + bias;
            float v1 = acc1[r] + bias;
            v0 = v0 > 0.f ? v0 : 0.f;                   // ReLU
            v1 = v1 > 0.f ? v1 : 0.f;
            sh_hid[r + half * 8][(wave * 2 + nt) * 16 + l16] = v0;
            sh_hid[16 + r + half * 8][(wave * 2 + nt) * 16 + l16] = v1;
        }
    }
    __syncthreads();

    // ---- GEMM2: out[32x64] = hidden[32x256] @ W2[256x64] ----
    // Wave -> (column tile, K half); B fragment shared by both M-subtiles,
    // so each W2 element is read exactly once per block.
    {
        const int nt2 = wave & 3;                       // output column tile
        const int kh  = wave >> 2;                      // K half (0..127 / 128..255)
        const int ncol = nt2 * 16 + l16;
        v8f acc0 = {}, acc1 = {};
        #pragma unroll 4
        for (int kk = 0; kk < 128; kk += 4) {
            const int kb = kh * 128 + kk + half * 2;
            v2f b;
            b.x = W2[(size_t)(kb + 0) * D_OUT + ncol];
            b.y = W2[(size_t)(kb + 1) * D_OUT + ncol];
            v2f a0, a1;
            a0.x = sh_hid[l16][kb + 0];
            a0.y = sh_hid[l16][kb + 1];
            a1.x = sh_hid[16 + l16][kb + 0];
            a1.y = sh_hid[16 + l16][kb + 1];
            acc0 = __builtin_amdgcn_wmma_f32_16x16x4_f32(
                false, a0, false, b, (short)0, acc0, false, false);
            acc1 = __builtin_amdgcn_wmma_f32_16x16x4_f32(
                false, a1, false, b, (short)0, acc1, false, false);
        }
        #pragma unroll
        for (int r = 0; r < 8; ++r) {
            sh_part[kh][r + half * 8][ncol]      = acc0[r];
            sh_part[kh][16 + r + half * 8][ncol] = acc1[r];
        }
    }
    __syncthreads();

    // ---- Reduce K-halves, add b2, write out (8 floats per thread) ----
    {
        int i = tid * 8;
        int m = i >> 6, f = i & 63;
        float4 o0, o1;
        o0.x = sh_part[0][m][f + 0] + sh_part[1][m][f + 0] + b2[f + 0];
        o0.y = sh_part[0][m][f + 1] + sh_part[1][m][f + 1] + b2[f + 1];
        o0.z = sh_part[0][m][f + 2] + sh_part[1][m][f + 2] + b2[f + 2];
        o0.w = sh_part[0][m][f + 3] + sh_part[1][m][f + 3] + b2[f + 3];
        o1.x = sh_part[0][m][f + 4] + sh_part[1][m][f + 4] + b2[f + 4];
        o1.y = sh_part[0][m][f + 5] + sh_part[1][m][f + 5] + b2[f + 5];
        o1.z = sh_part[0][m][f + 6] + sh_part[1][m][f + 6] + b2[f + 6];
        o1.w = sh_part[0][m][f + 7] + sh_part[1][m][f + 7] + b2[f + 7];
        size_t off = (size_t)(node0 + m) * D_OUT + f;
        *(float4*)(out + off)     = o0;
        *(float4*)(out + off + 4) = o1;
    }
}

// ---------------------------------------------------------------------------
extern "C" void kernel_launch(void* const* d_in, const int* in_sizes, int n_in,
                              void* d_out, int out_size, void* d_ws, size_t ws_size,
                              hipStream_t stream) {
    const float*     x  = (const float*)d_in[0];
    const long long* ei = (const long long*)d_in[1];   // int64 [2, N_EDGES]
    // d_in[2] = edge_attr (unused by the reference GINConv)
    const float* W1 = (const float*)d_in[3];
    const float* b1 = (const float*)d_in[4];
    const float* W2 = (const float*)d_in[5];
    const float* b2 = (const float*)d_in[6];
    float* out = (float*)d_out;

    const size_t aggBytes = (size_t)N_NODES * D_IN * sizeof(float);
    // Prefer workspace; falling back to d_out is safe: each MLP block reads
    // only its own rows of agg (into LDS, before any write) and later writes
    // exactly those rows of out.
    float* agg = (ws_size >= aggBytes) ? (float*)d_ws : out;

    // zero agg: N_NODES*D_IN / 4 floats per thread / 256 = 6250 blocks
    gin_zero<<<(N_NODES * D_IN) / (4 * 256), 256, 0, stream>>>(agg);

    // scatter: 16 threads per edge -> 16M threads -> 62500 blocks
    gin_scatter<<<(N_EDGES * 16) / 256, 256, 0, stream>>>(x, ei, agg);

    // fused MLP: one block per 32 nodes -> 3125 blocks
    gin_mlp<<<N_NODES / MT, 256, 0, stream>>>(x, agg, W1, b1, W2, b2, out);
}